// PGA_Unfold_JX_41772851921183
// MI455X (gfx1250) — compile-verified
//
#include <hip/hip_runtime.h>
#include <math.h>

typedef __attribute__((ext_vector_type(2))) float v2f;
typedef __attribute__((ext_vector_type(8))) float v8f;

#define LN2F 0.6931471805599453f

__device__ __forceinline__ float2 cmul(float2 a, float2 b){
  return make_float2(a.x*b.x - a.y*b.y, a.x*b.y + a.y*b.x);
}
__device__ __forceinline__ float2 cmulc(float2 a, float2 b){ // a * conj(b)
  return make_float2(a.x*b.x + a.y*b.y, a.y*b.x - a.x*b.y);
}
__device__ __forceinline__ float2 conjmul(float2 a, float2 b){ // conj(a) * b
  return make_float2(a.x*b.x + a.y*b.y, a.x*b.y - a.y*b.x);
}
__device__ __forceinline__ float2 cadd(float2 a, float2 b){ return make_float2(a.x+b.x, a.y+b.y); }
__device__ __forceinline__ float2 csub(float2 a, float2 b){ return make_float2(a.x-b.x, a.y-b.y); }
__device__ __forceinline__ float2 cscale(float2 a, float s){ return make_float2(a.x*s, a.y*s); }
__device__ __forceinline__ float2 cconj(float2 a){ return make_float2(a.x, -a.y); }

__device__ __forceinline__ float2 proj_um(float2 v){
  float mag = sqrtf(v.x*v.x + v.y*v.y);
  if (mag > 1e-12f) return make_float2(v.x/mag, v.y/mag);
  return make_float2(0.f, 0.f);
}

struct SM {
  float2 H[4][64];        // H[m][n]
  float2 F[64][8];        // F[n][r]
  float2 W[8][4];         // W[r][k]
  float2 t[64][4];        // t = F @ W
  float2 Mt[64][4];       // Mmat @ t
  float  MtD[2][64][8];   // WMMA real-composite staging
  float2 Bp[16][2][16];   // WMMA B panel; cols 8..15 zeroed once at start
  float2 g[4][8];
  float2 V[8][8];
  float2 a1[4][8];
  float2 c[4][8];
  float2 dm[4];
  float  inv1[4];
  float  inv2[4];
  float2 u[4][4];
  float  dW[4];
  float2 e[4][4];
  float  red[8];
};

// ---- block-wide sum: wave32 shfl tree + 8 per-wave partials in LDS ----
__device__ __forceinline__ float block_sum(SM& sm, float v){
  const int tid = threadIdx.x;
  #pragma unroll
  for (int off = 16; off > 0; off >>= 1) v += __shfl_xor(v, off, 32);
  if ((tid & 31) == 0) sm.red[tid >> 5] = v;
  __syncthreads();
  float r = 0.f;
  #pragma unroll
  for (int w = 0; w < 8; ++w) r += sm.red[w];
  __syncthreads();
  return r;
}

// ---- t = F @ W (64x4 complex), each thread owns one (n,k) ----
__device__ __forceinline__ void compute_t(SM& sm){
  const int tid = threadIdx.x;
  const int n = tid >> 2, k = tid & 3;
  float2 acc = make_float2(0.f, 0.f);
  #pragma unroll
  for (int r = 0; r < 8; ++r) acc = cadd(acc, cmul(sm.F[n][r], sm.W[r][k]));
  sm.t[n][k] = acc;
  __syncthreads();
}

__device__ __forceinline__ float power_t(SM& sm){
  const int tid = threadIdx.x;
  const int n = tid >> 2, k = tid & 3;
  float2 v = sm.t[n][k];
  return block_sum(sm, v.x*v.x + v.y*v.y);
}

__device__ __forceinline__ void scale_Ft(SM& sm, float s){
  const int tid = threadIdx.x;
  #pragma unroll
  for (int rep = 0; rep < 2; ++rep){
    int idx = tid + rep*256;
    sm.F[idx>>3][idx&7] = cscale(sm.F[idx>>3][idx&7], s);
  }
  const int n = tid >> 2, k = tid & 3;
  sm.t[n][k] = cscale(sm.t[n][k], s);
  __syncthreads();
}

// ---- g[m][r] = sum_n conj(F[n][r]) H[m][n]; 8 contiguous lanes per (m,r) ----
__device__ __forceinline__ void compute_g(SM& sm){
  const int tid = threadIdx.x;
  const int chunk = tid & 7, r = (tid >> 3) & 7, m = tid >> 6;
  float2 acc = make_float2(0.f, 0.f);
  #pragma unroll
  for (int q = 0; q < 8; ++q){
    int n = chunk*8 + q;
    acc = cadd(acc, conjmul(sm.F[n][r], sm.H[m][n]));
  }
  #pragma unroll
  for (int off = 1; off < 8; off <<= 1){
    acc.x += __shfl_xor(acc.x, off, 32);
    acc.y += __shfl_xor(acc.y, off, 32);
  }
  if (chunk == 0) sm.g[m][r] = acc;
  __syncthreads();
}

// ---- Mt = Mmat @ t via V_WMMA_F32_16X16X4_F32 (real composite) ----
// A slice lives permanently in registers (Areg, 16 x v2f per lane).
// Wave w: rt = w&3 (16-row tile), part = w>>2 (0: Re(M), 1: Im(M)).
// B panel cols 0..3 = Re(t) cols, 4..7 = Im(t) cols, 8..15 = persistent zeros.
__device__ __forceinline__ void mt_wmma(SM& sm, const v2f (&Areg)[16]){
  const int tid = threadIdx.x;
  // stage only live B columns: 256 entries, 1 per thread, branch-free select
  {
    int ks = tid >> 4, h = (tid >> 3) & 1, col = tid & 7;
    int row = ks*4 + h*2;
    int cc = col & 3;
    float2 t0 = sm.t[row][cc];
    float2 t1 = sm.t[row + 1][cc];
    float2 v;
    v.x = (col < 4) ? t0.x : t0.y;
    v.y = (col < 4) ? t1.x : t1.y;
    sm.Bp[ks][h][col] = v;
  }
  __syncthreads();
  const int lane = tid & 31;
  const int wave = tid >> 5;
  const int rt   = wave & 3;
  const int part = wave >> 2;
  const int half = lane >> 4;
  const int l15  = lane & 15;
  v8f acc = {0.f,0.f,0.f,0.f,0.f,0.f,0.f,0.f};
  #pragma unroll
  for (int ks = 0; ks < 16; ++ks){
    float2 bp = sm.Bp[ks][half][l15];
    v2f bb; bb[0] = bp.x; bb[1] = bp.y;
    acc = __builtin_amdgcn_wmma_f32_16x16x4_f32(false, Areg[ks], false, bb,
                                                (short)0, acc, false, false);
  }
  // D layout: VGPR v -> M = rt*16 + half*8 + v ; N = l15
  if (l15 < 8){
    #pragma unroll
    for (int v = 0; v < 8; ++v)
      sm.MtD[part][rt*16 + half*8 + v][l15] = acc[v];
  }
  __syncthreads();
  const int n = tid >> 2, k = tid & 3;
  float mr = sm.MtD[0][n][k]     - sm.MtD[1][n][k + 4];
  float mi = sm.MtD[0][n][k + 4] + sm.MtD[1][n][k];
  sm.Mt[n][k] = make_float2(mr, mi);
  __syncthreads();
}

__device__ __forceinline__ float trace_tMt(SM& sm){
  const int tid = threadIdx.x;
  const int n = tid >> 2, k = tid & 3;
  float2 a = sm.t[n][k], b = sm.Mt[n][k];
  return block_sum(sm, a.x*b.x + a.y*b.y);   // sum Re(conj(t) * Mt)
}

// ---- per-(m,s) coefficient of grad_F_com ----
__device__ __forceinline__ void grad_coeffs(SM& sm){
  const int tid = threadIdx.x;
  if (tid < 64){
    int r = tid >> 3, s = tid & 7;
    float2 acc = make_float2(0.f, 0.f);
    #pragma unroll
    for (int k = 0; k < 4; ++k) acc = cadd(acc, cmulc(sm.W[r][k], sm.W[s][k]));
    sm.V[r][s] = acc;
  }
  __syncthreads();
  if (tid < 32){
    int m = tid >> 3, s = tid & 7;
    float2 acc = make_float2(0.f, 0.f);
    #pragma unroll
    for (int r = 0; r < 8; ++r) acc = cadd(acc, conjmul(sm.g[m][r], sm.V[r][s]));
    sm.a1[m][s] = acc;
  }
  if (tid >= 32 && tid < 36){
    int m = tid - 32;
    float2 acc = make_float2(0.f, 0.f);
    #pragma unroll
    for (int r = 0; r < 8; ++r) acc = cadd(acc, conjmul(sm.g[m][r], sm.W[r][m]));
    sm.dm[m] = acc;
  }
  __syncthreads();
  if (tid < 4){
    int m = tid;
    float q1 = 0.f, q2 = 0.f;
    #pragma unroll
    for (int s = 0; s < 8; ++s){
      float2 A1 = sm.a1[m][s];
      float2 A2 = csub(A1, cmul(sm.dm[m], cconj(sm.W[s][m])));
      float2 gm = sm.g[m][s];
      q1 += A1.x*gm.x - A1.y*gm.y;
      q2 += A2.x*gm.x - A2.y*gm.y;
    }
    sm.inv1[m] = 1.f / (LN2F*(q1 + 1.f) + 1e-4f);
    sm.inv2[m] = 1.f / (LN2F*(q2 + 1.f) + 1e-4f);
  }
  __syncthreads();
  if (tid < 32){
    int m = tid >> 3, s = tid & 7;
    float2 A1 = sm.a1[m][s];
    float2 A2 = csub(A1, cmul(sm.dm[m], cconj(sm.W[s][m])));
    sm.c[m][s] = csub(cscale(A1, sm.inv1[m]), cscale(A2, sm.inv2[m]));
  }
  __syncthreads();
}

// ---- metrics (sum-rate, CRB, power) ----
__device__ __forceinline__ void record(SM& sm, const v2f (&Areg)[16], float* out,
                                       int b, int B, int R, int it, float xi0){
  const int tid = threadIdx.x;
  const int m = tid >> 6, k = (tid >> 4) & 3, chunk = tid & 15;
  float2 acc = make_float2(0.f, 0.f);
  #pragma unroll
  for (int q = 0; q < 4; ++q){
    int n = chunk*4 + q;
    acc = cadd(acc, conjmul(sm.H[m][n], sm.t[n][k]));
  }
  #pragma unroll
  for (int off = 1; off < 16; off <<= 1){
    acc.x += __shfl_xor(acc.x, off, 32);
    acc.y += __shfl_xor(acc.y, off, 32);
  }
  if (chunk == 0) sm.e[m][k] = acc;
  __syncthreads();
  mt_wmma(sm, Areg);
  float tr = trace_tMt(sm);
  float pw = power_t(sm);
  if (tid == 0){
    float rate = 0.f;
    #pragma unroll
    for (int mm = 0; mm < 4; ++mm){
      float sig = 0.f, tot = 0.f;
      #pragma unroll
      for (int kk = 0; kk < 4; ++kk){
        float2 ev = sm.e[mm][kk];
        float p = ev.x*ev.x + ev.y*ev.y;
        tot += p;
        if (kk == mm) sig = p;
      }
      float interf = tot - sig;
      rate += log2f(1.f + sig / (interf + 1.f));
    }
    out[(size_t)b*R + it] = rate;
    out[(size_t)B*R + (size_t)b*R + it] = xi0 / tr;
    out[2*(size_t)B*R + (size_t)b*R + it] = pw;
  }
  __syncthreads();
}

// ---- Mmat precompute: T = R_N_inv @ A_dot, Mmat = A_dot^H @ T ----
__global__ void mat_T(const float2* __restrict__ Rm, const float2* __restrict__ A,
                      float2* __restrict__ T){
  const int i = blockIdx.x, j = threadIdx.x;
  float2 acc = make_float2(0.f, 0.f);
  for (int k = 0; k < 64; ++k) acc = cadd(acc, cmul(Rm[i*64 + k], A[k*64 + j]));
  T[i*64 + j] = acc;
}
__global__ void mat_M(const float2* __restrict__ A, const float2* __restrict__ T,
                      float2* __restrict__ M){
  const int i = blockIdx.x, j = threadIdx.x;
  float2 acc = make_float2(0.f, 0.f);
  for (int k = 0; k < 64; ++k) acc = cadd(acc, conjmul(A[k*64 + i], T[k*64 + j]));
  M[i*64 + j] = acc;
}

// ---- main: one workgroup (256 threads = 8 waves) per batch element ----
__global__ __launch_bounds__(256)
void pga_kernel(const float2* __restrict__ H, const float2* __restrict__ F0,
                const float2* __restrict__ W0, const float* __restrict__ step,
                const float* __restrict__ xi0p, const float* __restrict__ Ptp,
                const int* __restrict__ noutp, const int* __restrict__ ninp,
                const float2* __restrict__ Mg, float* __restrict__ out, int B)
{
  __shared__ SM sm;
  const int tid = threadIdx.x;
  const int b = blockIdx.x;
  const int n_out = *noutp;
  const int n_in  = *ninp;
  const int R = n_out * (n_in + 1);
  const float xi0 = *xi0p;
  const float Pt  = *Ptp;

  // Preload this lane's invariant WMMA A-slice of Mmat into registers:
  // lanes 0-15: rows rt*16+l, K = ks*4+{0,1}; lanes 16-31: K = ks*4+{2,3};
  // waves 0-3 take Re(M), waves 4-7 take Im(M).
  v2f Areg[16];
  {
    const int lane = tid & 31, wave = tid >> 5;
    const int rt = wave & 3, part = wave >> 2;
    const int half = lane >> 4, l15 = lane & 15;
    const int arow = rt*16 + l15;
    #pragma unroll
    for (int ks = 0; ks < 16; ++ks){
      float2 m0 = Mg[arow*64 + ks*4 + half*2];
      float2 m1 = Mg[arow*64 + ks*4 + half*2 + 1];
      Areg[ks][0] = part ? m0.y : m0.x;
      Areg[ks][1] = part ? m1.y : m1.x;
    }
  }

  const float2* Hb = H  + (size_t)b*256;
  const float2* Fb = F0 + (size_t)b*512;
  const float2* Wb = W0 + (size_t)b*32;

  sm.H[tid >> 6][tid & 63] = Hb[tid];
  #pragma unroll
  for (int rep = 0; rep < 2; ++rep){
    int idx = tid + rep*256;
    sm.F[idx >> 3][idx & 7] = proj_um(Fb[idx]);   // initialize(): unit-modulus
  }
  if (tid < 32) sm.W[tid >> 2][tid & 3] = Wb[tid];
  // zero the dead B-panel columns (8..15) exactly once
  {
    int ks = tid >> 4, h = (tid >> 3) & 1, col8 = (tid & 7) + 8;
    sm.Bp[ks][h][col8] = make_float2(0.f, 0.f);
  }
  __syncthreads();

  compute_t(sm);
  {
    float P = power_t(sm);
    scale_Ft(sm, sqrtf(Pt / P));   // normalize_power_F
  }

  for (int ii = 0; ii < n_out; ++ii){
    for (int jj = 0; jj < n_in; ++jj){
      compute_g(sm);
      grad_coeffs(sm);
      mt_wmma(sm, Areg);                 // grad_F_crb numerator (pre-update t)
      float invtr = 1.f / trace_tMt(sm);
      float ss = step[(jj*n_out + ii)*2 + 0];
      // F += ss*gradFcom - 0.1*ss*gradFcrb
      #pragma unroll
      for (int rep = 0; rep < 2; ++rep){
        int idx = tid + rep*256;
        int n = idx >> 3, s = idx & 7;
        float2 gc = make_float2(0.f, 0.f);
        #pragma unroll
        for (int m = 0; m < 4; ++m) gc = cadd(gc, cmul(sm.H[m][n], sm.c[m][s]));
        float2 gb = make_float2(0.f, 0.f);
        #pragma unroll
        for (int k = 0; k < 4; ++k) gb = cadd(gb, cmulc(sm.Mt[n][k], sm.W[s][k]));
        float2 Fv = sm.F[n][s];
        Fv.x += ss*gc.x - 0.1f*ss*invtr*gb.x;
        Fv.y += ss*gc.y - 0.1f*ss*invtr*gb.y;
        sm.F[n][s] = Fv;
      }
      __syncthreads();
      compute_t(sm);
      {
        float P = power_t(sm);
        scale_Ft(sm, sqrtf(Pt / P));
      }
      record(sm, Areg, out, b, B, R, ii*(n_in + 1) + jj, xi0);
    }

    // ---- outer step: project F, update W ----
    #pragma unroll
    for (int rep = 0; rep < 2; ++rep){
      int idx = tid + rep*256;
      sm.F[idx >> 3][idx & 7] = proj_um(sm.F[idx >> 3][idx & 7]);
    }
    __syncthreads();
    compute_t(sm);
    compute_g(sm);
    if (tid < 16){
      int m = tid >> 2, k = tid & 3;
      float2 acc = make_float2(0.f, 0.f);
      #pragma unroll
      for (int r = 0; r < 8; ++r) acc = cadd(acc, conjmul(sm.g[m][r], sm.W[r][k]));
      sm.u[m][k] = acc;
    }
    __syncthreads();
    if (tid < 4){
      float ssum = 0.f;
      #pragma unroll
      for (int k = 0; k < 4; ++k){
        float2 uv = sm.u[tid][k];
        ssum += uv.x*uv.x + uv.y*uv.y;
      }
      sm.dW[tid] = 1.f / (LN2F*(ssum + 1.f));
    }
    __syncthreads();
    mt_wmma(sm, Areg);
    float invtr = 1.f / trace_tMt(sm);
    float ssw = step[ii*2 + 1];
    if (tid < 32){
      int r = tid >> 2, k = tid & 3;
      // grad_W_com eye-mask keeps m == k: g[k][r]*u[k][k]/denom[k]/K_users
      float2 gWc = cscale(cmul(sm.g[k][r], sm.u[k][k]), 0.25f * sm.dW[k]);
      float2 acc = make_float2(0.f, 0.f);
      for (int n = 0; n < 64; ++n) acc = cadd(acc, conjmul(sm.F[n][r], sm.Mt[n][k]));
      float2 gWb = cscale(acc, invtr);   // F^H M F W / tr
      float2 Wv = sm.W[r][k];
      Wv.x += ssw*gWc.x - 0.1f*ssw*gWb.x;
      Wv.y += ssw*gWc.y - 0.1f*ssw*gWb.y;
      sm.W[r][k] = Wv;
    }
    __syncthreads();
    compute_t(sm);
    {
      float P = power_t(sm);
      float s = sqrtf(Pt / P);
      if (tid < 32) sm.W[tid >> 2][tid & 3] = cscale(sm.W[tid >> 2][tid & 3], s);
      int n = tid >> 2, k = tid & 3;
      sm.t[n][k] = cscale(sm.t[n][k], s);
      __syncthreads();
    }
    record(sm, Areg, out, b, B, R, ii*(n_in + 1) + n_in, xi0);
  }

  // ---- final F, W outputs (complex64 interleaved) ----
  {
    float2* outF2 = reinterpret_cast<float2*>(out + 3*(size_t)B*R);
    #pragma unroll
    for (int rep = 0; rep < 2; ++rep){
      int idx = tid + rep*256;
      outF2[(size_t)b*512 + idx] = sm.F[idx >> 3][idx & 7];
    }
    float2* outW2 = reinterpret_cast<float2*>(out + 3*(size_t)B*R + (size_t)B*1024);
    if (tid < 32) outW2[(size_t)b*32 + tid] = sm.W[tid >> 2][tid & 3];
  }
}

extern "C" void kernel_launch(void* const* d_in, const int* in_sizes, int n_inputs,
                              void* d_out, int out_size, void* d_ws, size_t ws_size,
                              hipStream_t stream){
  (void)n_inputs; (void)out_size; (void)ws_size;
  const float2* H    = (const float2*)d_in[0];
  const float2* F0   = (const float2*)d_in[1];
  const float2* W0   = (const float2*)d_in[2];
  const float2* Adot = (const float2*)d_in[3];
  const float2* Rinv = (const float2*)d_in[4];
  const float*  step = (const float*)d_in[5];
  const float*  xi0  = (const float*)d_in[6];
  const float*  Pt   = (const float*)d_in[7];
  const int*    nout = (const int*)d_in[8];
  const int*    nin  = (const int*)d_in[9];

  float2* T = (float2*)d_ws;          // 64x64 scratch
  float2* M = T + 64*64;              // Mmat

  hipLaunchKernelGGL(mat_T, dim3(64), dim3(64), 0, stream, Rinv, Adot, T);
  hipLaunchKernelGGL(mat_M, dim3(64), dim3(64), 0, stream, Adot, T, M);

  const int B = in_sizes[0] / 256;    // H is (1,B,4,64) complex
  hipLaunchKernelGGL(pga_kernel, dim3(B), dim3(256), 0, stream,
                     H, F0, W0, step, xi0, Pt, nout, nin, M, (float*)d_out, B);
}